// LoraLayer_11965778887241
// MI455X (gfx1250) — compile-verified
//
#include <hip/hip_runtime.h>

// CDNA5 / gfx1250 fused LoRA kernel:  out = (x @ B) @ A * 2.0
// x: [M=8192, 4096] f32, B: [4096, 16] f32, A: [16, 4096] f32, out: [M, 4096] f32
// Uses V_WMMA_F32_16X16X4_F32 for both the rank-16 contraction and the expansion.

typedef __attribute__((ext_vector_type(2))) float v2f;
typedef __attribute__((ext_vector_type(8))) float v8f;

#define K_IN   4096
#define N_OUT  4096
#define RANK   16
#define WAVES  8          // 256 threads, wave32
#define MTILE  16         // rows of x per block

__global__ __launch_bounds__(256) void lora_fused_wmma(
    const float* __restrict__ x,     // [M, 4096]
    const float* __restrict__ lA,    // [16, 4096]
    const float* __restrict__ lB,    // [4096, 16]
    float* __restrict__ out)         // [M, 4096]
{
    // LDS: per-wave partial C tiles for the K-reduction + final xb tile
    __shared__ float red[WAVES][256];
    __shared__ float xb[16][17];     // padded to dodge bank conflicts

    const int tid  = threadIdx.x;
    const int wave = tid >> 5;       // 0..7 (wave32)
    const int lane = tid & 31;
    const int half = lane >> 4;      // 0: lanes 0-15, 1: lanes 16-31
    const int l15  = lane & 15;
    const size_t m0 = (size_t)blockIdx.x * MTILE;

    // ---------------- Phase 1: xb[16x16] = x_tile[16 x 4096] @ B[4096 x 16] ----------------
    // A-fragment (ISA 7.12.2, 32-bit A 16x4): lanes 0-15 hold row M=lane, K=k..k+1;
    // lanes 16-31 hold row M=lane-16, K=k+2..k+3  -> one contiguous b64 load per lane.
    v8f acc = {0.f, 0.f, 0.f, 0.f, 0.f, 0.f, 0.f, 0.f};

    const int kChunk = K_IN / WAVES;          // 512 K-values per wave
    const int kBase  = wave * kChunk;
    const float* xrow = x + (m0 + (size_t)l15) * K_IN;

    for (int k = kBase; k < kBase + kChunk; k += 4) {
        const int ka = k + half * 2;          // this lane's K pair
        v2f a;
        a.x = xrow[ka];
        a.y = xrow[ka + 1];
        // B-fragment (4x16, row striped across lanes within a VGPR):
        // VGPR0 = rows {k, k+2}, VGPR1 = rows {k+1, k+3} for the two half-waves.
        v2f b;
        b.x = lB[(size_t)ka * RANK + l15];
        b.y = lB[(size_t)(ka + 1) * RANK + l15];
        acc = __builtin_amdgcn_wmma_f32_16x16x4_f32(
            /*neg_a=*/false, a, /*neg_b=*/false, b,
            /*c_mod=*/(short)0, acc, /*reuse_a=*/false, /*reuse_b=*/false);
    }

    // Spill partial C tiles and reduce across the 8 waves.
#pragma unroll
    for (int r = 0; r < 8; ++r) red[wave][lane * 8 + r] = acc[r];
    __syncthreads();

    {
        const int l = tid >> 3;               // emulated lane 0..31
        const int r = tid & 7;                // emulated C VGPR 0..7
        float s = 0.f;
#pragma unroll
        for (int w = 0; w < WAVES; ++w) s += red[w][l * 8 + r];
        const int Mi = r + ((l >> 4) << 3);   // C layout: M = r (+8 for upper half)
        const int Ni = l & 15;                // N = lane
        xb[Mi][Ni] = 2.0f * s;                // fold in LoRA scale alpha/rank = 2
    }
    __syncthreads();

    // ---------------- Phase 2: out_tile[16 x 4096] = xb[16x16] @ A[16 x 4096] ----------------
    // Build the 4 A-fragments of xb (K=16 split into 4 steps of 4).
    v2f af[4];
#pragma unroll
    for (int t = 0; t < 4; ++t) {
        const int kk = 4 * t + half * 2;
        af[t].x = xb[l15][kk];
        af[t].y = xb[l15][kk + 1];
    }

    const int nTiles = N_OUT / 16;            // 256 tiles, 32 per wave
    for (int tile = wave; tile < nTiles; tile += WAVES) {
        const int n0 = tile * 16;
        v8f c = {0.f, 0.f, 0.f, 0.f, 0.f, 0.f, 0.f, 0.f};
#pragma unroll
        for (int t = 0; t < 4; ++t) {
            const int kk = 4 * t + half * 2;  // rows of lora_A for this fragment
            v2f b;
            b.x = lA[(size_t)kk * N_OUT + n0 + l15];
            b.y = lA[(size_t)(kk + 1) * N_OUT + n0 + l15];
            c = __builtin_amdgcn_wmma_f32_16x16x4_f32(
                false, af[t], false, b, (short)0, c, false, false);
        }
        // Store: C VGPR r -> row M = r + 8*half, col N = n0 + l15 (64B coalesced per half-wave)
#pragma unroll
        for (int r = 0; r < 8; ++r) {
            const size_t Mi = m0 + (size_t)(r + 8 * half);
            out[Mi * N_OUT + n0 + l15] = c[r];
        }
    }
}

extern "C" void kernel_launch(void* const* d_in, const int* in_sizes, int n_in,
                              void* d_out, int out_size, void* d_ws, size_t ws_size,
                              hipStream_t stream) {
    const float* x  = (const float*)d_in[0];   // [4, 2048, 4096]
    const float* lA = (const float*)d_in[1];   // [16, 4096]
    const float* lB = (const float*)d_in[2];   // [4096, 16]
    float* out = (float*)d_out;

    const int M = in_sizes[0] / K_IN;          // 8192 rows total
    const int blocks = M / MTILE;              // 512 workgroups
    lora_fused_wmma<<<dim3(blocks), dim3(256), 0, stream>>>(x, lA, lB, out);
}